// NW_2937757630656
// MI455X (gfx1250) — compile-verified
//
#include <hip/hip_runtime.h>
#include <math.h>

typedef float v2f __attribute__((ext_vector_type(2)));
typedef float v8f __attribute__((ext_vector_type(8)));

#define B_ROWS 4096
#define S_ROWS 8192
#define DIMS   16
#define NWAVES 8
#define TILES          (S_ROWS / 16)
#define TILES_PER_WAVE (TILES / NWAVES)

// ---------------- prologue: gather + interpolate + norms -------------------
__global__ __launch_bounds__(256) void nw_prep(const float* __restrict__ pairs,
                                               const int*   __restrict__ indices,
                                               const float* __restrict__ t,
                                               float* __restrict__ xt,
                                               float* __restrict__ p1,
                                               float* __restrict__ xtn) {
  int s = blockIdx.x * blockDim.x + threadIdx.x;
  if (s >= S_ROWS) return;
  const float tt  = t[0];
  const float omt = 1.0f - tt;
  const float* src = pairs + (size_t)indices[s] * (2 * DIMS);
  float nrm = 0.f;
#pragma unroll
  for (int d = 0; d < DIMS; ++d) {
    float a = src[d];
    float b = src[DIMS + d];
    float v = tt * b + omt * a;
    xt[(size_t)s * DIMS + d] = v;
    p1[(size_t)s * DIMS + d] = b;
    nrm += v * v;
  }
  xtn[s] = nrm;
}

// ---------------- fused flash-softmax attention with f32 WMMA --------------
// Per workgroup: one 16-row Q block. 8 waves split the S dimension; each wave
// runs online softmax over its tiles, accumulating O^T = V^T x P^T via WMMA.
__global__ __launch_bounds__(NWAVES * 32) void nw_attn(const float* __restrict__ xt,
                                                       const float* __restrict__ p1,
                                                       const float* __restrict__ xtn,
                                                       const float* __restrict__ x,
                                                       const float* __restrict__ t,
                                                       float* __restrict__ out) {
  __shared__ float ls_m[NWAVES][16];
  __shared__ float ls_l[NWAVES][16];
  __shared__ float ls_o[NWAVES][16][16];

  const int tid  = threadIdx.x;
  const int lane = tid & 31;
  const int w    = tid >> 5;
  const int h    = lane >> 4;   // wave half
  const int m    = lane & 15;   // fixed output row index for this lane
  const int qb   = blockIdx.x * 16;

  // pi(m) permutation so P^T registers land directly in B-operand layout
  const int npi = ((m & 7) >> 1) * 4 + (m & 1) + 2 * (m >> 3);

  // ---- Q block: B operand (row m as column m), plus ||q_m||^2 ----
  const float* qrow = x + (size_t)(qb + m) * DIMS;
  float qv[16];
#pragma unroll
  for (int i = 0; i < 4; ++i) {
    float4 q4 = ((const float4*)qrow)[i];
    qv[4 * i + 0] = q4.x; qv[4 * i + 1] = q4.y;
    qv[4 * i + 2] = q4.z; qv[4 * i + 3] = q4.w;
  }
  float qn = 0.f;
#pragma unroll
  for (int i = 0; i < 16; ++i) qn += qv[i] * qv[i];
  v2f bq[4];
#pragma unroll
  for (int c = 0; c < 4; ++c) {
    bq[c].x = qv[4 * c + 2 * h];
    bq[c].y = qv[4 * c + 2 * h + 1];
  }
  v2f bq4; bq4.x = h ? 0.0f : -0.5f; bq4.y = 0.0f;   // folds -||k||^2/2 into score

  float mrun = -3.0e38f, lrun = 0.0f;
  v8f  oacc = {0.f, 0.f, 0.f, 0.f, 0.f, 0.f, 0.f, 0.f};

  // Strided pointers: all per-tile loads become immediate-offset accesses.
  const int s0w = w * TILES_PER_WAVE * 16;             // wave's S base
  const float* xr  = xt  + (size_t)(s0w + npi) * DIMS + 2 * h;  // +1024/tile
  const float* knp = xtn + s0w + npi;                           // +16/tile
  const float* vb  = p1  + (size_t)(s0w + 2 * h) * DIMS + m;    // +1024/tile

  for (int tI = 0; tI < TILES_PER_WAVE; ++tI) {
    // A operand: x_t tile rows (permuted), K = 16 (+ norm chunk)
    v2f a0 = *(const v2f*)(xr + 0);
    v2f a1 = *(const v2f*)(xr + 4);
    v2f a2 = *(const v2f*)(xr + 8);
    v2f a3 = *(const v2f*)(xr + 12);
    float knv = *knp;
    v2f a4; a4.x = h ? 0.0f : knv; a4.y = 0.0f;

    // V^T operand elements: fixed immediate offsets from one base pointer
    float vv0[4], vv1[4];
#pragma unroll
    for (int c = 0; c < 4; ++c) {
      vv0[c] = vb[c * 64];        // row s0+4c+2h,   element d=m
      vv1[c] = vb[c * 64 + 16];   // row s0+4c+2h+1, element d=m
    }

    // S^T tile: st[r] = q_m . xt_n - 0.5*||xt_n||^2   (n in VGPR/half axis)
    v8f st = {0.f, 0.f, 0.f, 0.f, 0.f, 0.f, 0.f, 0.f};
    st = __builtin_amdgcn_wmma_f32_16x16x4_f32(false, a0, false, bq[0], (short)0, st, false, false);
    st = __builtin_amdgcn_wmma_f32_16x16x4_f32(false, a1, false, bq[1], (short)0, st, false, false);
    st = __builtin_amdgcn_wmma_f32_16x16x4_f32(false, a2, false, bq[2], (short)0, st, false, false);
    st = __builtin_amdgcn_wmma_f32_16x16x4_f32(false, a3, false, bq[3], (short)0, st, false, false);
    st = __builtin_amdgcn_wmma_f32_16x16x4_f32(false, a4, false, bq4,   (short)0, st, false, false);

    // score = -max(||q||^2 + ||k||^2 - 2 q.k, 0)/2 = min(st - 0.5*qn, 0)
    float sc[8];
    float tmax = -3.0e38f;
#pragma unroll
    for (int r = 0; r < 8; ++r) {
      sc[r] = fminf(st[r] - 0.5f * qn, 0.0f);
      tmax  = fmaxf(tmax, sc[r]);
    }
    tmax = fmaxf(tmax, __shfl_xor(tmax, 16, 32));   // other 8 n-values
    float mnew = fmaxf(mrun, tmax);

    float p[8];
    float ssum = 0.f;
#pragma unroll
    for (int r = 0; r < 8; ++r) {
      p[r] = __expf(sc[r] - mnew);
      ssum += p[r];
    }
    ssum += __shfl_xor(ssum, 16, 32);

    float corr = __expf(mrun - mnew);
    lrun = lrun * corr + ssum;
#pragma unroll
    for (int r = 0; r < 8; ++r) oacc[r] *= corr;
    mrun = mnew;

    // O^T += V^T x P^T ; thanks to pi(m), B chunks are just {p[2c], p[2c+1]}
#pragma unroll
    for (int c = 0; c < 4; ++c) {
      v2f av; av.x = vv0[c]; av.y = vv1[c];
      v2f bp; bp.x = p[2 * c]; bp.y = p[2 * c + 1];
      oacc = __builtin_amdgcn_wmma_f32_16x16x4_f32(false, av, false, bp, (short)0, oacc, false, false);
    }

    xr  += 16 * DIMS;   // next 16 S-rows
    knp += 16;
    vb  += 16 * DIMS;
  }

  // ---- merge the 8 waves' partial softmaxes (LSE merge) ----
  if (h == 0) { ls_m[w][m] = mrun; ls_l[w][m] = lrun; }
#pragma unroll
  for (int r = 0; r < 8; ++r) ls_o[w][r + 8 * h][m] = oacc[r];
  __syncthreads();

  if (w == 0) {
    float M = -3.0e38f;
#pragma unroll
    for (int ww = 0; ww < NWAVES; ++ww) M = fmaxf(M, ls_m[ww][m]);
    float L = 0.f;
    float om[8];
#pragma unroll
    for (int r = 0; r < 8; ++r) om[r] = 0.f;
#pragma unroll
    for (int ww = 0; ww < NWAVES; ++ww) {
      float sw = __expf(ls_m[ww][m] - M);
      L += sw * ls_l[ww][m];
#pragma unroll
      for (int r = 0; r < 8; ++r) om[r] += sw * ls_o[ww][r + 8 * h][m];
    }
    const float invL   = 1.0f / L;
    const float inv1mt = 1.0f / (1.0f - t[0]);
    // v = (softmax @ pi1 - x) / (1 - t)     [mean_s(omega) == 1 exactly]
#pragma unroll
    for (int r = 0; r < 8; ++r) {
      const int d = r + 8 * h;
      const size_t o = (size_t)(qb + m) * DIMS + d;
      out[o] = (om[r] * invL - x[o]) * inv1mt;
    }
  }
}

// ---------------------------------------------------------------------------
extern "C" void kernel_launch(void* const* d_in, const int* in_sizes, int n_in,
                              void* d_out, int out_size, void* d_ws, size_t ws_size,
                              hipStream_t stream) {
  (void)in_sizes; (void)n_in; (void)out_size; (void)ws_size;
  const float* x       = (const float*)d_in[0];
  const float* pairs   = (const float*)d_in[1];
  const float* t       = (const float*)d_in[2];
  const int*   indices = (const int*)d_in[3];
  float* out = (float*)d_out;

  float* xt  = (float*)d_ws;                 // [S,16]
  float* p1  = xt + (size_t)S_ROWS * DIMS;   // [S,16]
  float* xtn = p1 + (size_t)S_ROWS * DIMS;   // [S]

  nw_prep<<<(S_ROWS + 255) / 256, 256, 0, stream>>>(pairs, indices, t, xt, p1, xtn);
  nw_attn<<<B_ROWS / 16, NWAVES * 32, 0, stream>>>(xt, p1, xtn, x, t, out);
}